// NodeDot_21036749816029
// MI455X (gfx1250) — compile-verified
//
#include <hip/hip_runtime.h>

// CDNA5 / gfx1250, wave32.
// out[e] = dot(x[senders[e]], x[receivers[e]]), D_FEAT = 32.
//
// Strategy: one wave32 handles 16 edges. Sender rows form the A matrix
// (16x4 f32 slices), receiver rows form the B matrix (4x16), and
// V_WMMA_F32_16X16X4_F32 chained 8x over K=32 produces a 16x16 Gram tile
// whose DIAGONAL is the 16 edge dot products. Compute is free (kernel is
// L2-gather bound: 2.5M edges x 256B = 640MB of L2 row gathers; the 12.8MB
// node table is L2-resident on the 192MB L2), so the 16x MAC waste costs
// nothing and we keep full fp32 precision.

typedef __attribute__((ext_vector_type(2))) float v2f;
typedef __attribute__((ext_vector_type(8))) float v8f;

__global__ void __launch_bounds__(256) node_dot_wmma_kernel(
    const float* __restrict__ x,
    const int* __restrict__ senders,
    const int* __restrict__ receivers,
    float* __restrict__ out,
    int n_edges)
{
    const int lane = threadIdx.x & 31;   // wave32 lane
    const int m    = lane & 15;          // edge-in-group this lane serves
    const int h    = lane >> 4;          // K-half selector (ISA A/B layout)

    const int waves_per_block = blockDim.x >> 5;
    const int wave_id  = blockIdx.x * waves_per_block + (threadIdx.x >> 5);
    const int n_waves  = gridDim.x * waves_per_block;
    const int n_groups = (n_edges + 15) >> 4;

    // Uniform (lane-invariant) loop bounds => EXEC stays all-1s for WMMA.
    for (int g = wave_id; g < n_groups; g += n_waves) {
        int e = (g << 4) + m;
        int ec = e < n_edges ? e : n_edges - 1;   // clamp tail (loads only)

        // Index streams are touched once: non-temporal, don't pollute L2.
        int s = __builtin_nontemporal_load(senders + ec);
        int r = __builtin_nontemporal_load(receivers + ec);

        // Node rows: 128B each, L2-resident; regular temporal loads.
        const float* xs = x + (size_t)s * 32 + 2 * h;
        const float* xr = x + (size_t)r * 32 + 2 * h;

        v8f c = {};
#pragma unroll
        for (int k = 0; k < 8; ++k) {
            // ISA 16x4 f32 A layout: lane (h*16+m) holds (M=m, K=2h) in v0,
            // (M=m, K=2h+1) in v1 -> a natural 8-byte load per K-step.
            // B (4x16) mirrors it with N=m, receiver row.
            v2f a = *(const v2f*)(xs + 4 * k);
            v2f b = *(const v2f*)(xr + 4 * k);
            // 8 args: (neg_a, A, neg_b, B, c_mod, C, reuse_a, reuse_b)
            c = __builtin_amdgcn_wmma_f32_16x16x4_f32(
                    false, a, false, b, (short)0, c, false, false);
        }

        // Diagonal extraction from the 16x16 f32 C/D layout:
        //   lanes 0-15 : N=lane,    M = vgpr      -> diag m<8  at lane m,  c[m]
        //   lanes 16-31: N=lane-16, M = vgpr + 8  -> diag m>=8 at lane m+16, c[m-8]
        int sel = m & 7;
        float d = c[0];
        d = (sel == 1) ? c[1] : d;
        d = (sel == 2) ? c[2] : d;
        d = (sel == 3) ? c[3] : d;
        d = (sel == 4) ? c[4] : d;
        d = (sel == 5) ? c[5] : d;
        d = (sel == 6) ? c[6] : d;
        d = (sel == 7) ? c[7] : d;

        // Lane owns edge m's diagonal iff its half matches m's row half.
        if ((h == (m >> 3)) && (e < n_edges)) {
            // Output is streamed once: non-temporal store.
            __builtin_nontemporal_store(d, out + e);
        }
    }
}

extern "C" void kernel_launch(void* const* d_in, const int* in_sizes, int n_in,
                              void* d_out, int out_size, void* d_ws, size_t ws_size,
                              hipStream_t stream) {
    (void)n_in; (void)out_size; (void)d_ws; (void)ws_size;

    const float* x         = (const float*)d_in[0];
    const int*   senders   = (const int*)d_in[1];   // int32 per harness convention
    const int*   receivers = (const int*)d_in[2];
    float*       out       = (float*)d_out;

    const int n_edges  = in_sizes[1];               // 2,500,000
    const int n_groups = (n_edges + 15) / 16;       // 16 edges per wave32
    const int waves_per_block = 256 / 32;           // 8 waves/block on CDNA5
    int blocks = (n_groups + waves_per_block - 1) / waves_per_block;
    if (blocks < 1) blocks = 1;

    node_dot_wmma_kernel<<<blocks, 256, 0, stream>>>(x, senders, receivers,
                                                     out, n_edges);
}